// CustomActivation_46385646797497
// MI455X (gfx1250) — compile-verified
//
#include <hip/hip_runtime.h>

// Elementwise 3-level threshold activation, 8192x8192 fp32.
// Memory-bound: 512 MB traffic @ 23.3 TB/s -> ~22 us floor on MI455X.
// Each thread: 16 consecutive floats = 4 consecutive b128 NT loads sharing one
// base address (immediate offsets 0/16/32/48), then 4 b128 NT stores.
// A wave's 4 load instructions cover one contiguous, fully-utilized 2KB span.
// NT hints because the 512MB stream exceeds the 192MB L2 (avoid thrash).

typedef __attribute__((ext_vector_type(4))) float v4f;

#define T1  0.33f
#define T2  0.66f
#define LO  0.333333333f
#define MID 0.6666666666f
#define HI  1.0f

__device__ __forceinline__ float quant1(float x) {
    // two v_cmp + two v_cndmask, branchless, bit-exact vs jnp.where chain
    return x < T1 ? LO : (x < T2 ? MID : HI);
}

__device__ __forceinline__ v4f quant4(v4f x) {
    v4f r;
    r.x = quant1(x.x);
    r.y = quant1(x.y);
    r.z = quant1(x.z);
    r.w = quant1(x.w);
    return r;
}

__global__ void __launch_bounds__(256)
CustomActivation_46385646797497_kernel(const float* __restrict__ in,
                                       float* __restrict__ out,
                                       unsigned int n) {
    const v4f* __restrict__ vin  = (const v4f*)in;
    v4f* __restrict__       vout = (v4f*)out;

    const unsigned int n4   = n >> 2;                               // float4 count
    const unsigned int t    = blockIdx.x * blockDim.x + threadIdx.x;
    const unsigned int base = t << 2;                               // first float4 idx

    if (base + 3 < n4) {
        // Fast path: 4 consecutive b128 NT loads off one base address.
        v4f a = __builtin_nontemporal_load(&vin[base + 0]);
        v4f b = __builtin_nontemporal_load(&vin[base + 1]);
        v4f c = __builtin_nontemporal_load(&vin[base + 2]);
        v4f d = __builtin_nontemporal_load(&vin[base + 3]);
        __builtin_nontemporal_store(quant4(a), &vout[base + 0]);
        __builtin_nontemporal_store(quant4(b), &vout[base + 1]);
        __builtin_nontemporal_store(quant4(c), &vout[base + 2]);
        __builtin_nontemporal_store(quant4(d), &vout[base + 3]);
    } else {
        // Ragged-edge vectors (none for 8192*8192: n4 % 4 == 0)
        for (unsigned int j = base; j < n4; ++j) {
            v4f a = __builtin_nontemporal_load(&vin[j]);
            __builtin_nontemporal_store(quant4(a), &vout[j]);
        }
    }

    // Scalar tail for n % 4 elements (empty for 8192*8192)
    const unsigned int scalarBase = n4 << 2;
    if (t < (n & 3u)) {
        out[scalarBase + t] = quant1(in[scalarBase + t]);
    }
}

extern "C" void kernel_launch(void* const* d_in, const int* in_sizes, int n_in,
                              void* d_out, int out_size, void* d_ws, size_t ws_size,
                              hipStream_t stream) {
    (void)in_sizes; (void)n_in; (void)d_ws; (void)ws_size;

    const float* in  = (const float*)d_in[0];
    float*       out = (float*)d_out;
    const unsigned int n = (unsigned int)out_size;   // 8192*8192 = 67,108,864

    const int block = 256;                           // 8 wave32 waves per block
    const unsigned int n4 = n >> 2;
    // one thread per 4 float4s (16 floats)
    long long threads_needed = ((long long)n4 + 3) / 4;
    if (threads_needed < 1) threads_needed = 1;
    long long blocks = (threads_needed + block - 1) / block;

    CustomActivation_46385646797497_kernel<<<(int)blocks, block, 0, stream>>>(in, out, n);
}